// UIR_KG_5111011082891
// MI455X (gfx1250) — compile-verified
//
#include <hip/hip_runtime.h>
#include <hip/hip_bf16.h>
#include <math.h>

typedef __attribute__((ext_vector_type(2))) float v2f;
typedef __attribute__((ext_vector_type(8))) float v8f;

#define OUT_STRIDE 112
#define EMBED_DIM 64

// ---------------------------------------------------------------------------
// Fills
// ---------------------------------------------------------------------------
__global__ void zero_f4_kernel(float* __restrict__ p, long long n4) {
    long long gid = (long long)blockIdx.x * blockDim.x + threadIdx.x;
    if (gid < n4) ((float4*)p)[gid] = make_float4(0.f, 0.f, 0.f, 0.f);
}
__global__ void zero_i_kernel(int* __restrict__ p, int n) {
    int gid = blockIdx.x * blockDim.x + threadIdx.x;
    if (gid < n) p[gid] = 0;
}

// ---------------------------------------------------------------------------
// out[:, 0:64] = E   (row-major strided copy, float4 per thread)
// ---------------------------------------------------------------------------
__global__ void copy_embed_kernel(const float* __restrict__ E,
                                  float* __restrict__ out, int n_nodes) {
    long long gid = (long long)blockIdx.x * blockDim.x + threadIdx.x;
    long long total = (long long)n_nodes * (EMBED_DIM / 4);
    if (gid >= total) return;
    int row = (int)(gid >> 4);
    int c4  = (int)(gid & 15);
    float4 v = ((const float4*)E)[gid];
    *(float4*)(out + (size_t)row * OUT_STRIDE + c4 * 4) = v;
}

// ---------------------------------------------------------------------------
// CSR build: histogram -> exclusive scan -> slot reorder of (col, val)
// ---------------------------------------------------------------------------
__global__ void hist_kernel(const int* __restrict__ rows, int* __restrict__ cnt,
                            int n_edges) {
    int e = blockIdx.x * blockDim.x + threadIdx.x;
    if (e < n_edges) atomicAdd(&cnt[rows[e]], 1);
}

// Per-block exclusive scan over chunks of 1024 counts (256 thr x 4 elems).
__global__ __launch_bounds__(256) void scan_block_kernel(
    const int* __restrict__ counts, int* __restrict__ out,
    int* __restrict__ blocksums, int n) {
    __shared__ int lds[256];
    const int t = threadIdx.x;
    const int base = blockIdx.x * 1024;
    int v[4]; int sum = 0;
#pragma unroll
    for (int i = 0; i < 4; ++i) {
        int idx = base + t * 4 + i;
        v[i] = (idx < n) ? counts[idx] : 0;
        sum += v[i];
    }
    lds[t] = sum;
    __syncthreads();
    for (int ofs = 1; ofs < 256; ofs <<= 1) {       // Hillis-Steele inclusive
        int x = (t >= ofs) ? lds[t - ofs] : 0;
        __syncthreads();
        lds[t] += x;
        __syncthreads();
    }
    int run = (t == 0) ? 0 : lds[t - 1];
#pragma unroll
    for (int i = 0; i < 4; ++i) {
        int idx = base + t * 4 + i;
        if (idx < n) out[idx] = run;
        run += v[i];
    }
    if (t == 255) blocksums[blockIdx.x] = lds[255];
}

__global__ void scan_top_kernel(int* __restrict__ blocksums, int nb) {
    if (blockIdx.x == 0 && threadIdx.x == 0) {
        int run = 0;
        for (int i = 0; i < nb; ++i) { int v = blocksums[i]; blocksums[i] = run; run += v; }
    }
}

__global__ void scan_add_kernel(int* __restrict__ row_ptr,
                                const int* __restrict__ blocksums,
                                int n, int n_edges) {
    int idx = blockIdx.x * blockDim.x + threadIdx.x;
    if (idx < n) row_ptr[idx] += blocksums[idx >> 10];
    if (idx == 0) row_ptr[n] = n_edges;
}

__global__ void reorder_kernel(const int* __restrict__ rows,
                               const int* __restrict__ cols,
                               const float* __restrict__ vals,
                               const int* __restrict__ row_ptr,
                               int* __restrict__ fill,
                               int* __restrict__ pcol, float* __restrict__ pval,
                               int n_edges) {
    int e = blockIdx.x * blockDim.x + threadIdx.x;
    if (e >= n_edges) return;
    int r = rows[e];
    int pos = row_ptr[r] + atomicAdd(&fill[r], 1);
    pcol[pos] = cols[e];
    pval[pos] = vals[e];
}

// ---------------------------------------------------------------------------
// Pull-model SpMM: side[r,:] = sum_{j in row r} pval[j] * x[pcol[j],:]
// One wave per row; edge metadata fetched 32-wide then lane-broadcast (shfl);
// x-gathers are coalesced 128/256B lines (L2-resident). No atomics.
// ---------------------------------------------------------------------------
template <int D>
__global__ __launch_bounds__(256) void pull_side_kernel(
    const float* __restrict__ x, const int* __restrict__ row_ptr,
    const int* __restrict__ pcol, const float* __restrict__ pval,
    float* __restrict__ side, int n_rows) {
    constexpr int PER = D / 32;       // floats per lane (2 or 1)
    const int wave = threadIdx.x >> 5;
    const int lane = threadIdx.x & 31;
    const int r = blockIdx.x * 8 + wave;
    if (r >= n_rows) return;
    const int beg = row_ptr[r], end = row_ptr[r + 1];

    float acc0 = 0.f, acc1 = 0.f;
    for (int j0 = beg; j0 < end; j0 += 32) {
        int   myj = j0 + lane;
        int   c_l = (myj < end) ? pcol[myj] : 0;
        float v_l = (myj < end) ? pval[myj] : 0.f;
        __builtin_prefetch((const void*)(pcol + j0 + 256), 0, 1);
        __builtin_prefetch((const void*)(pval + j0 + 256), 0, 1);
        const int cnt = min(32, end - j0);
        for (int t = 0; t < cnt; ++t) {
            int   c = __shfl(c_l, t, 32);
            float v = __shfl(v_l, t, 32);
            if (PER == 2) {
                float2 xv = ((const float2*)(x + (size_t)c * D))[lane];
                acc0 += v * xv.x;
                acc1 += v * xv.y;
            } else {
                acc0 += v * x[(size_t)c * D + lane];
            }
        }
    }
    if (PER == 2) {
        float2 o; o.x = acc0; o.y = acc1;
        ((float2*)(side + (size_t)r * D))[lane] = o;
    } else {
        side[(size_t)r * D + lane] = acc0;
    }
}

// ---------------------------------------------------------------------------
// Fallback push scatter (used only if ws_size can't hold the CSR buffers).
// ---------------------------------------------------------------------------
template <int D>
__global__ __launch_bounds__(256) void scatter_kernel(
    const float* __restrict__ x, const float* __restrict__ vals,
    const int* __restrict__ rows, const int* __restrict__ cols,
    float* __restrict__ side, int n_edges) {
    constexpr int C = D / 4;
    constexpr int LOGC = (D == 64) ? 4 : 3;
    long long total  = (long long)n_edges * C;
    long long stride = (long long)gridDim.x * blockDim.x;
    for (long long gid = (long long)blockIdx.x * blockDim.x + threadIdx.x;
         gid < total; gid += stride) {
        int e = (int)(gid >> LOGC);
        int c = (int)(gid & (C - 1));
        float v  = vals[e];
        int   cl = cols[e];
        int   rw = rows[e];
        float4 xv = *(const float4*)(x + (size_t)cl * D + c * 4);
        float* sp = side + (size_t)rw * D + c * 4;
        atomicAdd(sp + 0, v * xv.x);
        atomicAdd(sp + 1, v * xv.y);
        atomicAdd(sp + 2, v * xv.z);
        atomicAdd(sp + 3, v * xv.w);
    }
}

// ---------------------------------------------------------------------------
// KGAT bi-interaction layer via fp32 WMMA (V_WMMA_F32_16X16X4_F32).
// Weights staged in LDS once per block -> K-loop feeds WMMA from ds_load_b64.
// One wave per 16-node tile; 8 waves/block; spare waves clamp to last tile so
// EXEC is all-ones for every WMMA and barriers are uniform.
// ---------------------------------------------------------------------------
template <int DIN, int DOUT>
__global__ __launch_bounds__(256) void transform_kernel(
    const float* __restrict__ x, const float* __restrict__ side,
    const float* __restrict__ W1, const float* __restrict__ b1,
    const float* __restrict__ W2, const float* __restrict__ b2,
    float* __restrict__ x_next,       // [N, DOUT] un-normalized (next layer in)
    float* __restrict__ out,          // d_out + column offset, stride OUT_STRIDE
    int n_tiles) {
    constexpr int NT = DOUT / 16;
    __shared__ float lds_w1[DOUT * DIN];
    __shared__ float lds_w2[DOUT * DIN];
    __shared__ float lds_b[2 * DOUT];
    __shared__ float lds_y[8][16 * DOUT];

    // Cooperative weight staging (whole block).
    for (int i = threadIdx.x; i < DOUT * DIN; i += 256) {
        lds_w1[i] = W1[i];
        lds_w2[i] = W2[i];
    }
    for (int i = threadIdx.x; i < DOUT; i += 256) {
        lds_b[i] = b1[i];
        lds_b[DOUT + i] = b2[i];
    }
    __syncthreads();

    const int wave  = threadIdx.x >> 5;
    const int lane  = threadIdx.x & 31;
    int tile = blockIdx.x * 8 + wave;
    if (tile >= n_tiles) tile = n_tiles - 1;   // keep wave live, EXEC full
    const int mlane = lane & 15;
    const int khalf = lane >> 4;               // 0: K pair {0,1}, 1: {2,3}
    const int row   = tile * 16 + mlane;

    v8f csum[NT], cbi[NT];
#pragma unroll
    for (int nt = 0; nt < NT; ++nt)
#pragma unroll
        for (int i = 0; i < 8; ++i) { csum[nt][i] = 0.f; cbi[nt][i] = 0.f; }

    const float* xr = x    + (size_t)row * DIN;
    const float* sr = side + (size_t)row * DIN;

#pragma unroll 4
    for (int k0 = 0; k0 < DIN; k0 += 4) {
        const int kk = k0 + khalf * 2;
        float x0 = xr[kk], x1v = xr[kk + 1];
        float s0 = sr[kk], s1v = sr[kk + 1];
        v2f asum = {x0 + s0, x1v + s1v};   // (x + side) row fragment
        v2f abi  = {x0 * s0, x1v * s1v};   // (x * side) row fragment
#pragma unroll
        for (int nt = 0; nt < NT; ++nt) {
            const int n = nt * 16 + mlane;
            const float* w1r = lds_w1 + n * DIN + kk;   // B[k][n] = W[n][k]
            const float* w2r = lds_w2 + n * DIN + kk;
            v2f bf1 = {w1r[0], w1r[1]};
            v2f bf2 = {w2r[0], w2r[1]};
            csum[nt] = __builtin_amdgcn_wmma_f32_16x16x4_f32(
                false, asum, false, bf1, (short)0, csum[nt], false, false);
            cbi[nt]  = __builtin_amdgcn_wmma_f32_16x16x4_f32(
                false, abi,  false, bf2, (short)0, cbi[nt],  false, false);
        }
    }

    // Epilogue: bias + LeakyReLU(0.01) + combine -> per-wave LDS tile
    float* myy = &lds_y[wave][0];
#pragma unroll
    for (int nt = 0; nt < NT; ++nt) {
        const int n = nt * 16 + mlane;
        const float bb1 = lds_b[n], bb2 = lds_b[DOUT + n];
#pragma unroll
        for (int v = 0; v < 8; ++v) {
            const int m = v + khalf * 8;
            float a = csum[nt][v] + bb1; a = (a > 0.f) ? a : 0.01f * a;
            float c = cbi[nt][v]  + bb2; c = (c > 0.f) ? c : 0.01f * c;
            myy[m * DOUT + n] = a + c;
        }
    }
    __syncthreads();

    // Row-wise L2 normalize; lanes 0..15 each own one node row.
    if (lane < 16) {
        const int node = tile * 16 + lane;
        const float* yr = &lds_y[wave][lane * DOUT];
        float ss = 0.f;
#pragma unroll
        for (int n = 0; n < DOUT; ++n) { float t = yr[n]; ss += t * t; }
        const float inv = 1.f / fmaxf(sqrtf(ss), 1e-12f);
        float* xn = x_next + (size_t)node * DOUT;
        float* op = out    + (size_t)node * OUT_STRIDE;
#pragma unroll
        for (int n = 0; n < DOUT; ++n) {
            float t = yr[n];
            xn[n] = t;
            op[n] = t * inv;
        }
    }
}

// ---------------------------------------------------------------------------
extern "C" void kernel_launch(void* const* d_in, const int* in_sizes, int n_in,
                              void* d_out, int out_size, void* d_ws, size_t ws_size,
                              hipStream_t stream) {
    const float* E    = (const float*)d_in[0];
    const float* vals = (const float*)d_in[1];
    const int*   rows = (const int*)d_in[2];
    const int*   cols = (const int*)d_in[3];
    const float* W1_0 = (const float*)d_in[4];
    const float* b1_0 = (const float*)d_in[5];
    const float* W2_0 = (const float*)d_in[6];
    const float* b2_0 = (const float*)d_in[7];
    const float* W1_1 = (const float*)d_in[8];
    const float* b1_1 = (const float*)d_in[9];
    const float* W2_1 = (const float*)d_in[10];
    const float* b2_1 = (const float*)d_in[11];
    float* out = (float*)d_out;

    const int n_nodes = in_sizes[0] / EMBED_DIM;
    const int n_edges = in_sizes[1];
    const int n_tiles = (n_nodes + 15) / 16;

    // Workspace layout
    float* side = (float*)d_ws;                        // n*64 (reused at n*32)
    float* x1   = side + (size_t)n_nodes * 64;         // n*32
    float* x2   = x1   + (size_t)n_nodes * 32;         // n*16
    int*   row_ptr = (int*)(x2 + (size_t)n_nodes * 16);// n+1
    int*   fill    = row_ptr + (n_nodes + 1);          // n
    int*   bsums   = fill + n_nodes;                   // scan block sums (<=1024)
    int*   pcol    = bsums + 1024;                     // n_edges
    float* pval    = (float*)(pcol + n_edges);         // n_edges
    const size_t need_csr = (size_t)((char*)(pval + n_edges) - (char*)d_ws);
    const bool use_csr = ws_size >= need_csr;

    const int eb = (n_edges + 255) / 256;
    const int nb_scan = (n_nodes + 1023) / 1024;

    // out[:, 0:64] = E
    {
        long long t = (long long)n_nodes * (EMBED_DIM / 4);
        copy_embed_kernel<<<(int)((t + 255) / 256), 256, 0, stream>>>(E, out, n_nodes);
    }

    if (use_csr) {
        // ---- CSR build (graph identical for both layers) ----
        zero_i_kernel<<<(n_nodes + 255) / 256, 256, 0, stream>>>(fill, n_nodes);
        hist_kernel<<<eb, 256, 0, stream>>>(rows, fill, n_edges);
        scan_block_kernel<<<nb_scan, 256, 0, stream>>>(fill, row_ptr, bsums, n_nodes);
        scan_top_kernel<<<1, 32, 0, stream>>>(bsums, nb_scan);
        scan_add_kernel<<<(n_nodes + 255) / 256, 256, 0, stream>>>(row_ptr, bsums,
                                                                   n_nodes, n_edges);
        zero_i_kernel<<<(n_nodes + 255) / 256, 256, 0, stream>>>(fill, n_nodes);
        reorder_kernel<<<eb, 256, 0, stream>>>(rows, cols, vals, row_ptr, fill,
                                               pcol, pval, n_edges);

        // ---- Layer 0: 64 -> 32 (atomic-free pull SpMM) ----
        pull_side_kernel<64><<<(n_nodes + 7) / 8, 256, 0, stream>>>(
            E, row_ptr, pcol, pval, side, n_nodes);
        transform_kernel<64, 32><<<(n_tiles + 7) / 8, 256, 0, stream>>>(
            E, side, W1_0, b1_0, W2_0, b2_0, x1, out + 64, n_tiles);

        // ---- Layer 1: 32 -> 16 ----
        pull_side_kernel<32><<<(n_nodes + 7) / 8, 256, 0, stream>>>(
            x1, row_ptr, pcol, pval, side, n_nodes);
        transform_kernel<32, 16><<<(n_tiles + 7) / 8, 256, 0, stream>>>(
            x1, side, W1_1, b1_1, W2_1, b2_1, x2, out + 96, n_tiles);
    } else {
        // ---- Fallback: push scatter with fp32 atomics ----
        {
            long long n4 = (long long)n_nodes * 64 / 4;
            zero_f4_kernel<<<(int)((n4 + 255) / 256), 256, 0, stream>>>(side, n4);
            long long total = (long long)n_edges * 16;
            scatter_kernel<64><<<(int)((total + 255) / 256), 256, 0, stream>>>(
                E, vals, rows, cols, side, n_edges);
            transform_kernel<64, 32><<<(n_tiles + 7) / 8, 256, 0, stream>>>(
                E, side, W1_0, b1_0, W2_0, b2_0, x1, out + 64, n_tiles);
        }
        {
            long long n4 = (long long)n_nodes * 32 / 4;
            zero_f4_kernel<<<(int)((n4 + 255) / 256), 256, 0, stream>>>(side, n4);
            long long total = (long long)n_edges * 8;
            scatter_kernel<32><<<(int)((total + 255) / 256), 256, 0, stream>>>(
                x1, vals, rows, cols, side, n_edges);
            transform_kernel<32, 16><<<(n_tiles + 7) / 8, 256, 0, stream>>>(
                x1, side, W1_1, b1_1, W2_1, b2_1, x2, out + 96, n_tiles);
        }
    }
}